// Decoder_71837622993526
// MI455X (gfx1250) — compile-verified
//
#include <hip/hip_runtime.h>
#include <hip/hip_bf16.h>

typedef __attribute__((ext_vector_type(2))) float v2f;
typedef __attribute__((ext_vector_type(8))) float v8f;

#if __has_builtin(__builtin_amdgcn_global_load_async_to_lds_b128) && \
    __has_builtin(__builtin_amdgcn_s_wait_asynccnt)
#define USE_ASYNC_LDS 1
#else
#define USE_ASYNC_LDS 0
#endif

#if USE_ASYNC_LDS
typedef int v4i __attribute__((vector_size(16)));
typedef __attribute__((address_space(1))) v4i* gv4i_p;   // global (AS1)
typedef __attribute__((address_space(3))) v4i* lv4i_p;   // LDS (AS3)
#endif

// ---------------------------------------------------------------------------
// Generic fp32 WMMA GEMM:  out(64,N) = act( A0(64,1024)@W0(N,1024)^T
//                                          [+ A1(64,1024)@W1(N,1024)^T]
//                                          + bias0 [+ bias1] )
// A0 may be gathered through idx (embedding rows).  Block = 128 threads
// (4 waves); each wave owns a 64x16 output tile (full M), so each W row is
// streamed from HBM exactly once device-wide.  K-chunks of 32 are staged in
// double-buffered LDS via GLOBAL_LOAD_ASYNC_TO_LDS_B128 (ASYNCcnt) so the
// next chunk's HBM->LDS copy overlaps the current chunk's WMMA work.
// Matrix core: V_WMMA_F32_16X16X4_F32 (wave32).
// LDS row pitch 36 floats: 16B-aligned b128 stores and conflict-free
// ds_load_2addr_b32 fragment reads (36*n mod 64 distinct over both halves).
// ---------------------------------------------------------------------------
#define GK 1024
#define KC 32
#define LDSROW 36

__global__ __launch_bounds__(128) void gemm_wmma_f32(
    const float* __restrict__ A0, const int* __restrict__ idx,
    const float* __restrict__ W0,
    const float* __restrict__ A1, const float* __restrict__ W1,
    const float* __restrict__ bias0, const float* __restrict__ bias1,
    float* __restrict__ out, int N, int act /*0=none,1=tanh*/)
{
    __shared__ __align__(16) float As[2][64 * LDSROW];
    __shared__ __align__(16) float Ws[2][4][16 * LDSROW];

    const int tid    = threadIdx.x;
    const int lane   = tid & 31;
    const int wave   = tid >> 5;
    const int n_wave = blockIdx.x * 64 + wave * 16;

    const int nchunks = (A1 != nullptr ? 2 : 1) * (GK / KC);

    // Stage one 64xKC A-chunk + this wave's 16xKC W-chunk into LDS buffer b.
    auto stage = [&](int c, int b) {
        const int  csrc   = c >> 5;            // 32 chunks per source
        const int  k0     = (c & 31) * KC;
        const bool gather = (csrc == 0) && (idx != nullptr);
        const float* A = csrc ? A1 : A0;
        const float* W = csrc ? W1 : W0;
        float* AsB = &As[b][0];
        float* WsB = &Ws[b][wave][0];
        const float* Wp = W + (size_t)n_wave * GK + k0;
#if USE_ASYNC_LDS
        #pragma unroll
        for (int i = 0; i < 4; ++i) {          // A: 512 float4 / 128 thr
            int q = tid + 128 * i;
            int m = q >> 3, c4 = q & 7;
            int row = gather ? idx[m] : m;
            const float* gp = A + (size_t)row * GK + k0 + c4 * 4;
            float*       lp = AsB + m * LDSROW + c4 * 4;
            __builtin_amdgcn_global_load_async_to_lds_b128(
                (gv4i_p)(size_t)gp, (lv4i_p)lp, 0, 0);
        }
        #pragma unroll
        for (int i = 0; i < 4; ++i) {          // W: 128 float4 / 32 lanes
            int q = lane + 32 * i;
            int r = q >> 3, c4 = q & 7;
            const float* gp = Wp + (size_t)r * GK + c4 * 4;
            float*       lp = WsB + r * LDSROW + c4 * 4;
            __builtin_amdgcn_global_load_async_to_lds_b128(
                (gv4i_p)(size_t)gp, (lv4i_p)lp, 0, 0);
        }
#else
        #pragma unroll
        for (int i = 0; i < 4; ++i) {
            int q = tid + 128 * i;
            int m = q >> 3, c4 = q & 7;
            int row = gather ? idx[m] : m;
            float4 v = *(const float4*)(A + (size_t)row * GK + k0 + c4 * 4);
            *(float4*)(AsB + m * LDSROW + c4 * 4) = v;
        }
        #pragma unroll
        for (int i = 0; i < 4; ++i) {
            int q = lane + 32 * i;
            int r = q >> 3, c4 = q & 7;
            float4 v = *(const float4*)(Wp + (size_t)r * GK + c4 * 4);
            *(float4*)(WsB + r * LDSROW + c4 * 4) = v;
        }
#endif
    };

    v8f acc[4];
    #pragma unroll
    for (int ms = 0; ms < 4; ++ms) acc[ms] = (v8f)(0.0f);

    stage(0, 0);                               // prologue: fill buffer 0

    const int koff = (lane >> 4) * 2;          // lanes 0-15: K{0,1}; 16-31: K{2,3}
    const int nl   = lane & 15;

    for (int c = 0; c < nchunks; ++c) {
        const int b = c & 1;
#if USE_ASYNC_LDS
        __builtin_amdgcn_s_wait_asynccnt(0);   // this wave's copies landed
#endif
        __syncthreads();                       // all waves' chunk-c data visible,
                                               // and all reads of buffer b^1 done
        if (c + 1 < nchunks) stage(c + 1, b ^ 1);

        const float* AsB = &As[b][0];
        const float* WsB = &Ws[b][wave][0];
        #pragma unroll
        for (int kk = 0; kk < KC; kk += 4) {
            v2f bf;                            // B 4x16: B[k][n] = W[n][k]
            bf.x = WsB[nl * LDSROW + kk + koff];
            bf.y = WsB[nl * LDSROW + kk + koff + 1];
            #pragma unroll
            for (int ms = 0; ms < 4; ++ms) {
                v2f af;                        // A 16x4 fragment
                af.x = AsB[(ms * 16 + nl) * LDSROW + kk + koff];
                af.y = AsB[(ms * 16 + nl) * LDSROW + kk + koff + 1];
                acc[ms] = __builtin_amdgcn_wmma_f32_16x16x4_f32(
                    false, af, false, bf, (short)0, acc[ms], false, false);
            }
        }
    }

    // Epilogue: C/D layout -> VGPR r: lanes 0-15 = M r, lanes 16-31 = M r+8.
    const int mo = (lane >> 4) * 8;
    const int n  = n_wave + nl;
    float bb = bias0[n];
    if (bias1 != nullptr) bb += bias1[n];
    #pragma unroll
    for (int ms = 0; ms < 4; ++ms) {
        #pragma unroll
        for (int r = 0; r < 8; ++r) {
            int m = ms * 16 + r + mo;
            float v = acc[ms][r] + bb;
            if (act == 1) v = tanhf(v);
            out[(size_t)m * N + n] = v;
        }
    }
}

// ---------------------------------------------------------------------------
// LSTM pointwise: gates(64,4096) [i|f|g|o] -> h_new, c_new (64,1024)
// ---------------------------------------------------------------------------
__global__ __launch_bounds__(256) void lstm_pointwise(
    const float* __restrict__ gates, const float* __restrict__ c0,
    float* __restrict__ h_new, float* __restrict__ c_new)
{
    int i = blockIdx.x * 256 + threadIdx.x;    // 0 .. 65535
    int b = i >> 10, h = i & 1023;
    const float* g = gates + (size_t)b * 4096;
    float gi = g[h], gf = g[h + 1024], gg = g[h + 2048], go = g[h + 3072];
    float si = 1.0f / (1.0f + expf(-gi));
    float sf = 1.0f / (1.0f + expf(-gf));
    float so = 1.0f / (1.0f + expf(-go));
    float c  = sf * c0[i] + si * tanhf(gg);
    float hv = so * tanhf(c);
    c_new[i] = c;
    h_new[i] = hv;
}

// ---------------------------------------------------------------------------
// Attention: one block per batch row. scores over S=128, softmax, context t.
// ---------------------------------------------------------------------------
__global__ __launch_bounds__(128) void attention_kernel(
    const float* __restrict__ h_new, const float* __restrict__ enc,
    float* __restrict__ tvec)
{
    int b = blockIdx.x;
    int s = threadIdx.x;                       // 0..127
    __shared__ float red[128];
    __shared__ __align__(16) float alpha[128];
    const float* eb = enc + (size_t)b * 1024 * 128;
    const float* hb = h_new + (size_t)b * 1024;

    float sc = 0.0f;
    for (int h = 0; h < 1024; ++h)
        sc = fmaf(hb[h], eb[(size_t)h * 128 + s], sc);

    red[s] = sc; __syncthreads();
    for (int off = 64; off > 0; off >>= 1) {
        if (s < off) red[s] = fmaxf(red[s], red[s + off]);
        __syncthreads();
    }
    float mx = red[0]; __syncthreads();
    float e = expf(sc - mx);
    red[s] = e; __syncthreads();
    for (int off = 64; off > 0; off >>= 1) {
        if (s < off) red[s] += red[s + off];
        __syncthreads();
    }
    float inv = 1.0f / red[0];
    alpha[s] = e * inv;
    __syncthreads();

    for (int h = s; h < 1024; h += 128) {
        const float4* row = (const float4*)(eb + (size_t)h * 128);
        const float4* al  = (const float4*)alpha;
        float acc = 0.0f;
        #pragma unroll 8
        for (int ss = 0; ss < 32; ++ss) {
            float4 r4 = row[ss], a4 = al[ss];
            acc = fmaf(a4.x, r4.x, acc);
            acc = fmaf(a4.y, r4.y, acc);
            acc = fmaf(a4.z, r4.z, acc);
            acc = fmaf(a4.w, r4.w, acc);
        }
        tvec[(size_t)b * 1024 + h] = acc;
    }
}

// ---------------------------------------------------------------------------
// In-place row log_softmax over V=32000 (one block per batch row).
// ---------------------------------------------------------------------------
__global__ __launch_bounds__(256) void logsoftmax_kernel(float* __restrict__ y)
{
    int b = blockIdx.x;
    int t = threadIdx.x;
    float* row = y + (size_t)b * 32000;
    __shared__ float red[256];

    float mx = -INFINITY;
    for (int v = t; v < 32000; v += 256) mx = fmaxf(mx, row[v]);
    red[t] = mx; __syncthreads();
    for (int off = 128; off > 0; off >>= 1) {
        if (t < off) red[t] = fmaxf(red[t], red[t + off]);
        __syncthreads();
    }
    mx = red[0]; __syncthreads();

    float sum = 0.0f;
    for (int v = t; v < 32000; v += 256) sum += expf(row[v] - mx);
    red[t] = sum; __syncthreads();
    for (int off = 128; off > 0; off >>= 1) {
        if (t < off) red[t] += red[t + off];
        __syncthreads();
    }
    float lse = mx + logf(red[0]);

    for (int v = t; v < 32000; v += 256) row[v] -= lse;
}

// ---------------------------------------------------------------------------
extern "C" void kernel_launch(void* const* d_in, const int* in_sizes, int n_in,
                              void* d_out, int out_size, void* d_ws, size_t ws_size,
                              hipStream_t stream)
{
    const int*   prev = (const int*)  d_in[0];
    const float* h0   = (const float*)d_in[1];
    const float* c0   = (const float*)d_in[2];
    const float* enc  = (const float*)d_in[3];
    const float* emb  = (const float*)d_in[4];
    const float* W_ih = (const float*)d_in[5];
    const float* W_hh = (const float*)d_in[6];
    const float* b_ih = (const float*)d_in[7];
    const float* b_hh = (const float*)d_in[8];
    const float* W1_w = (const float*)d_in[9];
    const float* W1_b = (const float*)d_in[10];
    const float* W2_w = (const float*)d_in[11];
    const float* W2_b = (const float*)d_in[12];
    const float* W_w  = (const float*)d_in[13];
    const float* W_b  = (const float*)d_in[14];

    float* pred  = (float*)d_out;                       // (64, 32000)
    float* h_new = pred  + (size_t)64 * 32000;          // (64, 1024)
    float* c_new = h_new + (size_t)64 * 1024;           // (64, 1024)

    float* gates = (float*)d_ws;                        // (64, 4096)
    float* tvec  = gates + (size_t)64 * 4096;           // (64, 1024)
    float* h_att = tvec  + (size_t)64 * 1024;           // (64, 1024)

    // 1) LSTM gates: emb[prev] @ W_ih^T + h0 @ W_hh^T + b_ih + b_hh
    gemm_wmma_f32<<<4096 / 64, 128, 0, stream>>>(
        emb, prev, W_ih, h0, W_hh, b_ih, b_hh, gates, 4096, 0);

    // 2) LSTM pointwise -> h_new, c_new (directly into d_out)
    lstm_pointwise<<<64 * 1024 / 256, 256, 0, stream>>>(gates, c0, h_new, c_new);

    // 3) Attention context t
    attention_kernel<<<64, 128, 0, stream>>>(h_new, enc, tvec);

    // 4) h_att = tanh(h_new @ W1^T + t @ W2^T + W1_b + W2_b)
    gemm_wmma_f32<<<1024 / 64, 128, 0, stream>>>(
        h_new, nullptr, W1_w, tvec, W2_w, W1_b, W2_b, h_att, 1024, 1);

    // 5) y = h_att @ W_w^T + W_b   (into pred region)
    gemm_wmma_f32<<<32000 / 64, 128, 0, stream>>>(
        h_att, nullptr, W_w, nullptr, nullptr, W_b, nullptr, pred, 32000, 0);

    // 6) pred = log_softmax(y) in place
    logsoftmax_kernel<<<64, 256, 0, stream>>>(pred);
}